// EMA_40707700032404
// MI455X (gfx1250) — compile-verified
//
#include <hip/hip_runtime.h>

// EMA along T: y_t = a*y_{t-1} + alpha*x_t, y_0 = x_0.
// x: [B=8, T=8192, C=512] f32. Chunked scan (16 chunks of 512 steps):
//   P1: per-chunk local scan end values (f64)
//   P2: carry scan across chunks (f64)
//   P3: scan-as-matmul via V_WMMA_F32_16X16X4_F32, 16-step tiles, f32

#define ALPHA_  0.01
#define AVAL_   0.99
#define B_      8
#define T_      8192
#define C_      512
#define CL_     512
#define NCH_    16

typedef __attribute__((ext_vector_type(2))) float v2f;
typedef __attribute__((ext_vector_type(8))) float v8f;

// ---------------- Phase 1: local chunk scans (f64) ----------------
__global__ void ema_p1(const float* __restrict__ x, double* __restrict__ E) {
    int tid = blockIdx.x * blockDim.x + threadIdx.x;   // 65536 threads
    int c = tid & (C_ - 1);
    int k = (tid >> 9) & (NCH_ - 1);
    int b = tid >> 13;
    const float* p = x + ((size_t)b * T_ + (size_t)k * CL_) * C_ + c;
    double y = 0.0;
    int t = 0;
    if (k == 0) { y = (double)p[0]; t = 1; }           // y_0 = x_0 special case
    for (; t < CL_; ++t)
        y = AVAL_ * y + ALPHA_ * (double)p[(size_t)t * C_];
    E[((size_t)b * NCH_ + k) * C_ + c] = y;
}

// ---------------- Phase 2: carry scan across chunks (f64) ----------------
__global__ void ema_p2(const double* __restrict__ E, double* __restrict__ carry) {
    int tid = blockIdx.x * blockDim.x + threadIdx.x;   // 4096 threads
    int c = tid & (C_ - 1);
    int b = tid >> 9;
    // a^512 by repeated squaring
    double a512 = AVAL_;
    #pragma unroll
    for (int i = 0; i < 9; ++i) a512 *= a512;
    double F = 0.0;
    for (int k = 0; k < NCH_; ++k) {
        size_t idx = ((size_t)b * NCH_ + k) * C_ + c;
        carry[idx] = F;                                 // carry-in for chunk k
        F = E[idx] + a512 * F;
    }
}

// ---------------- Phase 3: WMMA scan-as-matmul ----------------
// One wave handles one (b, chunk, 16-channel tile); loops over 32 tiles of
// 16 timesteps. Y(16x16) = W @ X via 4x wmma_f32_16x16x4_f32, then add
// a^{r+1}*carry per row; carry chained via shuffle from row 15.
__global__ void ema_p3(const float* __restrict__ x,
                       const double* __restrict__ carry,
                       float* __restrict__ out) {
    const int lane = threadIdx.x & 31;
    const int wav  = threadIdx.x >> 5;
    const int gw   = blockIdx.x * (blockDim.x >> 5) + wav;   // 4096 waves
    const int ct   = gw & 31;            // channel tile (16 channels)
    const int k    = (gw >> 5) & 15;     // chunk
    const int b    = gw >> 9;            // batch
    const int n    = lane & 15;          // column (B/C/D) and row (A)
    const int half = lane >> 4;
    const int c0   = ct * 16;

    // a^0 .. a^16
    float ap[17];
    ap[0] = 1.0f;
    #pragma unroll
    for (int i = 1; i <= 16; ++i) ap[i] = ap[i - 1] * (float)AVAL_;

    // A operand: W[m][kk] = (kk<=m) ? alpha*a^(m-kk) : 0, m = lane&15.
    // 16x4 f32 A layout: VGPR0 = K=kk0 (lanes 0-15) / K=kk0+2 (lanes 16-31),
    // VGPR1 = K=kk0+1 / K=kk0+3  ->  per-lane kk base = 4j + half*2.
    v2f Aop[4];
    #pragma unroll
    for (int j = 0; j < 4; ++j) {
        int kk = 4 * j + half * 2;
        Aop[j].x = (kk     <= n) ? (float)ALPHA_ * ap[n - kk]     : 0.0f;
        Aop[j].y = (kk + 1 <= n) ? (float)ALPHA_ * ap[n - kk - 1] : 0.0f;
    }

    // running carry (full EMA value just before this chunk) for this lane's column
    float crun = (float)carry[((size_t)b * NCH_ + k) * C_ + c0 + n];

    const size_t baseT = (size_t)b * T_;
    for (int tt = 0; tt < CL_ / 16; ++tt) {
        const int t0 = k * CL_ + tt * 16;
        v8f acc = {};
        #pragma unroll
        for (int j = 0; j < 4; ++j) {
            // B operand 4x16: lanes 0-15 rows K=4j,4j+1; lanes 16-31 rows K=4j+2,4j+3
            const int tr = 4 * j + half * 2;
            const float* p = x + (baseT + (size_t)(t0 + tr)) * C_ + c0 + n;
            float b0 = p[0];
            float b1 = p[C_];
            // global t==0: weight is a^r, not alpha*a^r -> feed x0/alpha
            b0 = (t0 + tr == 0) ? b0 * (float)(1.0 / ALPHA_) : b0;
            v2f Bop; Bop.x = b0; Bop.y = b1;
            acc = __builtin_amdgcn_wmma_f32_16x16x4_f32(
                false, Aop[j], false, Bop, (short)0, acc, false, false);
        }
        // add carry term: row r = v + half*8 gets a^{r+1} * crun
        #pragma unroll
        for (int v = 0; v < 8; ++v)
            acc[v] += ap[v + half * 8 + 1] * crun;
        // store: lane holds column n, rows v + half*8
        float* o = out + (baseT + (size_t)(t0 + half * 8)) * C_ + c0 + n;
        #pragma unroll
        for (int v = 0; v < 8; ++v)
            o[(size_t)v * C_] = acc[v];
        // next carry = full y at row 15: VGPR7, lanes 16-31 (column = lane-16)
        crun = __shfl(acc[7], n + 16, 32);
    }
}

extern "C" void kernel_launch(void* const* d_in, const int* in_sizes, int n_in,
                              void* d_out, int out_size, void* d_ws, size_t ws_size,
                              hipStream_t stream) {
    const float* x = (const float*)d_in[0];
    float* out = (float*)d_out;
    double* E     = (double*)d_ws;                       // 8*16*512 f64 = 512 KB
    double* carry = E + (size_t)B_ * NCH_ * C_;          // another 512 KB

    ema_p1<<<(B_ * NCH_ * C_) / 256, 256, 0, stream>>>(x, E);
    ema_p2<<<(B_ * C_) / 256, 256, 0, stream>>>(E, carry);
    ema_p3<<<(B_ * NCH_ * (C_ / 16) * 32) / 256, 256, 0, stream>>>(x, carry, out);
}